// MoEFeedForward_19739669692845
// MI455X (gfx1250) — compile-verified
//
#include <hip/hip_runtime.h>
#include <hip/hip_bf16.h>

#define DIMC 1024
#define HID  2816
#define NEXP 8
#define NTOK 8192
#define TM   16     // tokens per workgroup
#define HT   128    // H chunk
#define NWAVE 8     // 256 threads

typedef float v2f __attribute__((ext_vector_type(2)));
typedef float v8f __attribute__((ext_vector_type(8)));

// ---------------- Router: logits -> softmax -> top2 -> combine + partial sums
__global__ __launch_bounds__(256)
void moe_router_kernel(const float* __restrict__ x,
                       const float* __restrict__ gate_w,
                       float* __restrict__ combine,
                       float* __restrict__ partials) {
    __shared__ float gws[NEXP * DIMC];      // 32 KB
    __shared__ float red[256 * 16];         // 16 KB
    const int tid = threadIdx.x;
    for (int i = tid; i < NEXP * DIMC; i += 256) gws[i] = gate_w[i];
    __syncthreads();

    const int t = blockIdx.x * 256 + tid;
    const float* xr = x + (size_t)t * DIMC;

    float lg[NEXP];
    #pragma unroll
    for (int e = 0; e < NEXP; ++e) lg[e] = 0.f;
    for (int c = 0; c < DIMC; ++c) {
        float xv = xr[c];
        #pragma unroll
        for (int e = 0; e < NEXP; ++e) lg[e] = fmaf(xv, gws[e * DIMC + c], lg[e]);
    }
    // softmax
    float m = lg[0];
    #pragma unroll
    for (int e = 1; e < NEXP; ++e) m = fmaxf(m, lg[e]);
    float p[NEXP], s = 0.f;
    #pragma unroll
    for (int e = 0; e < NEXP; ++e) { p[e] = __expf(lg[e] - m); s += p[e]; }
    float inv = 1.f / s;
    #pragma unroll
    for (int e = 0; e < NEXP; ++e) p[e] *= inv;
    // top-2 (first index wins ties, like jax top_k)
    int i0 = 0; float p0 = p[0];
    #pragma unroll
    for (int e = 1; e < NEXP; ++e) if (p[e] > p0) { p0 = p[e]; i0 = e; }
    int i1 = (i0 == 0) ? 1 : 0; float p1 = p[i1];
    #pragma unroll
    for (int e = 0; e < NEXP; ++e)
        if (e != i0 && p[e] > p1) { p1 = p[e]; i1 = e; }
    float sn = 1.f / (p0 + p1);
    #pragma unroll
    for (int e = 0; e < NEXP; ++e)
        combine[(size_t)t * NEXP + e] = (e == i0) ? p0 * sn : ((e == i1) ? p1 * sn : 0.f);

    // deterministic block reduction of probs + expert mask
    #pragma unroll
    for (int e = 0; e < NEXP; ++e) {
        red[tid * 16 + e]     = p[e];
        red[tid * 16 + 8 + e] = (e == i0 || e == i1) ? 1.f : 0.f;
    }
    __syncthreads();
    for (int st = 128; st > 0; st >>= 1) {
        if (tid < st)
            #pragma unroll
            for (int v = 0; v < 16; ++v) red[tid * 16 + v] += red[(tid + st) * 16 + v];
        __syncthreads();
    }
    if (tid < 16) partials[blockIdx.x * 16 + tid] = red[tid];
}

__global__ void moe_aux_kernel(const float* __restrict__ partials, float* __restrict__ out_aux) {
    if (threadIdx.x == 0) {
        float sp[NEXP], sm[NEXP];
        #pragma unroll
        for (int e = 0; e < NEXP; ++e) { sp[e] = 0.f; sm[e] = 0.f; }
        for (int b = 0; b < NTOK / 256; ++b)
            #pragma unroll
            for (int e = 0; e < NEXP; ++e) {
                sp[e] += partials[b * 16 + e];
                sm[e] += partials[b * 16 + 8 + e];
            }
        const float invN = 1.f / (float)NTOK;
        float aux = 0.f;
        #pragma unroll
        for (int e = 0; e < NEXP; ++e) aux += (sp[e] * invN) * (sm[e] * invN);
        *out_aux = (float)NEXP * aux;
    }
}

// ---------------- Fused SwiGLU MoE FFN, f32 WMMA 16x16x4
__global__ __launch_bounds__(256)
void moe_ffn_kernel(const float* __restrict__ x,
                    const float* __restrict__ w1,
                    const float* __restrict__ w2,
                    const float* __restrict__ w3,
                    const float* __restrict__ combine,
                    float* __restrict__ out) {
    __shared__ float xT[DIMC * TM];  // x transposed: [c][tok]  64 KB
    __shared__ float hT[HT * TM];    // h transposed: [hcol][tok] 8 KB

    const int tid  = threadIdx.x;
    const int wave = tid >> 5;
    const int lane = tid & 31;
    const int lm   = lane & 15;      // A-row (m) / B-col (n) / D-col (token)
    const int half = lane >> 4;      // K half-select
    const int tokbase = blockIdx.x * TM;

    // cooperative transposed load of the x tile
    for (int i = tid; i < TM * DIMC; i += 256) {
        int t = i >> 10;             // DIMC == 1024
        int c = i & (DIMC - 1);
        xT[c * TM + t] = x[(size_t)(tokbase + t) * DIMC + c];
    }
    __syncthreads();

    // persistent y^T accumulators: 8 tiles of 16(C-rows) x 16(tokens)
    v8f acc[8];
    v8f zf = {};
    #pragma unroll
    for (int j = 0; j < 8; ++j) acc[j] = zf;

    for (int e = 0; e < NEXP; ++e) {
        const float cw = combine[(size_t)(tokbase + lm) * NEXP + e];
        if (!__any(cw != 0.f)) continue;   // tile uses no tokens of this expert

        const float* w1e = w1 + (size_t)e * HID * DIMC;
        const float* w3e = w3 + (size_t)e * HID * DIMC;
        const float* w2e = w2 + (size_t)e * DIMC * HID;

        for (int hb = 0; hb < HID; hb += HT) {
            // ---- phase A: h^T[hb+wave*16 .. +16][tok] = silu(W1·x^T) * (W3·x^T) * cw
            const float* a1p = w1e + (size_t)(hb + wave * 16 + lm) * DIMC + half * 2;
            const float* a3p = w3e + (size_t)(hb + wave * 16 + lm) * DIMC + half * 2;
            v8f d1 = zf, d3 = zf;
            #pragma unroll 4
            for (int k = 0; k < DIMC; k += 4) {
                v2f a1 = *(const v2f*)(a1p + k);
                v2f a3 = *(const v2f*)(a3p + k);
                const int kk = k + half * 2;
                v2f b;
                b.x = xT[kk * TM + lm];
                b.y = xT[(kk + 1) * TM + lm];
                d1 = __builtin_amdgcn_wmma_f32_16x16x4_f32(false, a1, false, b,
                                                           (short)0, d1, false, false);
                d3 = __builtin_amdgcn_wmma_f32_16x16x4_f32(false, a3, false, b,
                                                           (short)0, d3, false, false);
            }
            // h = silu(d1) * d3 * cw; sigmoid via hardware v_rcp_f32 (no IEEE div chain)
            #pragma unroll
            for (int v = 0; v < 8; ++v) {
                float s1  = d1[v];
                float sig = __builtin_amdgcn_rcpf(1.f + __expf(-s1));
                float hv  = s1 * sig * d3[v] * cw;
                int hloc  = wave * 16 + v + half * 8;    // hcol within chunk
                hT[hloc * TM + lm] = hv;
            }
            __syncthreads();

            // ---- phase B: y^T[wave*128 .. +128][tok] += W2chunk · h^T
            #pragma unroll 2
            for (int k = 0; k < HT; k += 4) {
                const int kk = k + half * 2;
                v2f b;
                b.x = hT[kk * TM + lm];
                b.y = hT[(kk + 1) * TM + lm];
                #pragma unroll
                for (int j = 0; j < 8; ++j) {
                    const int crow = wave * 128 + j * 16 + lm;
                    v2f a = *(const v2f*)(w2e + (size_t)crow * HID + hb + k + half * 2);
                    acc[j] = __builtin_amdgcn_wmma_f32_16x16x4_f32(false, a, false, b,
                                                                   (short)0, acc[j], false, false);
                }
            }
            __syncthreads();   // protect hT before next chunk rewrites it
        }
    }

    // ---- epilogue: scatter y^T tiles to out[token][c]
    #pragma unroll
    for (int j = 0; j < 8; ++j) {
        const int cbase = wave * 128 + j * 16;
        #pragma unroll
        for (int v = 0; v < 8; ++v) {
            const int crow = cbase + v + half * 8;
            out[(size_t)(tokbase + lm) * DIMC + crow] = acc[j][v];
        }
    }
}

extern "C" void kernel_launch(void* const* d_in, const int* in_sizes, int n_in,
                              void* d_out, int out_size, void* d_ws, size_t ws_size,
                              hipStream_t stream) {
    const float* x      = (const float*)d_in[0];
    const float* gate_w = (const float*)d_in[1];
    const float* w1     = (const float*)d_in[2];
    const float* w2     = (const float*)d_in[3];
    const float* w3     = (const float*)d_in[4];
    float* out = (float*)d_out;

    float* combine  = (float*)d_ws;                  // N*8 floats
    float* partials = combine + (size_t)NTOK * NEXP; // 32 blocks * 16 floats

    moe_router_kernel<<<NTOK / 256, 256, 0, stream>>>(x, gate_w, combine, partials);
    moe_aux_kernel<<<1, 32, 0, stream>>>(partials, out + (size_t)NTOK * DIMC);
    moe_ffn_kernel<<<NTOK / TM, 256, 0, stream>>>(x, w1, w2, w3, combine, out);
}